// HGT_360777253417
// MI455X (gfx1250) — compile-verified
//
#include <hip/hip_runtime.h>
#include <math.h>

// ---------------- problem constants (match reference) ----------------
#define NVERT 50000
#define NEDGE 10000
#define NPAIR 800000
#define CIN   128
#define CH    512      // HEADS * C_HID
#define HEADS 8
#define NCLS  40
#define SLOPEC 0.2f
#define EPS   1e-9f
#define NEGINF_ENC 0x007FFFFFu   // fflip(-inf)

typedef float v2f __attribute__((ext_vector_type(2)));
typedef float v8f __attribute__((ext_vector_type(8)));

static __device__ __forceinline__ float leaky(float x){ return x > 0.f ? x : SLOPEC * x; }

// monotonic unsigned encoding of float for atomicMax-based segment max
static __device__ __forceinline__ unsigned fflip(float f){
  unsigned u = __float_as_uint(f);
  return (u & 0x80000000u) ? ~u : (u | 0x80000000u);
}
static __device__ __forceinline__ float funflip(unsigned u){
  u = (u & 0x80000000u) ? (u ^ 0x80000000u) : ~u;
  return __uint_as_float(u);
}

static __device__ __forceinline__ float wred32(float t){
#pragma unroll
  for (int m = 16; m > 0; m >>= 1) t += __shfl_xor(t, m, 32);
  return t;
}

// ---------------- generic fill ----------------
__global__ void k_fill(unsigned* __restrict__ p, unsigned v, long long n){
  long long i = (long long)blockIdx.x * blockDim.x + threadIdx.x;
  long long st = (long long)gridDim.x * blockDim.x;
  for (; i < n; i += st) p[i] = v;
}

// ---------------- GEMM1: Xp[v, h*64+c] = sum_k X[v,k]*Wh[h,k,c] + bh[h,c] ----------------
// f32 WMMA 16x16x4, one 16x16 tile per wave. M=50000 (3125 tiles), N=512 (32 tiles), K=128.
__global__ void k_gemm1(const float* __restrict__ X, const float* __restrict__ Wh,
                        const float* __restrict__ bh, float* __restrict__ Xp){
  const int lane = threadIdx.x & 31;
  const int wave = threadIdx.x >> 5;
  const int NT = CH / 16;                       // 32 col tiles
  int tile = blockIdx.x * 8 + wave;
  if (tile >= (NVERT / 16) * NT) return;
  int tm = tile / NT, tn = tile % NT;
  int row0 = tm * 16, col0 = tn * 16;
  int half = lane >> 4, lr = lane & 15;
  int m = row0 + lr;                            // A row handled by this lane
  int n = col0 + lr;                            // B/C column handled by this lane
  int h = n >> 6, c = n & 63;
  const float* Arow  = X  + (long long)m * CIN;
  const float* Bbase = Wh + (long long)h * CIN * 64 + c;   // element (k,n) = Bbase[k*64]
  v8f acc = {};
  for (int k0 = 0; k0 < CIN; k0 += 4){
    int k = k0 + 2 * half;
    v2f a; a.x = Arow[k];               a.y = Arow[k + 1];
    v2f b; b.x = Bbase[(long long)k * 64]; b.y = Bbase[(long long)(k + 1) * 64];
    acc = __builtin_amdgcn_wmma_f32_16x16x4_f32(false, a, false, b, (short)0, acc, false, false);
  }
  float bias = bh[n];
#pragma unroll
  for (int r = 0; r < 8; r++){
    int row = row0 + r + 8 * half;              // C/D layout: VGPR r -> M=r (lo half) / r+8 (hi half)
    Xp[(long long)row * CH + n] = acc[r] + bias;
  }
}

// ---------------- GEMM2: Xp2 = Xt @ Wo + bo   (M=50000, K=512, N=40 padded to 48) --------
__global__ void k_gemm2(const float* __restrict__ Xt, const float* __restrict__ Wo,
                        const float* __restrict__ bo, float* __restrict__ Xp2){
  const int lane = threadIdx.x & 31;
  const int wave = threadIdx.x >> 5;
  const int NT = 3;
  int tile = blockIdx.x * 8 + wave;
  if (tile >= (NVERT / 16) * NT) return;
  int tm = tile / NT, tn = tile % NT;
  int row0 = tm * 16, col0 = tn * 16;
  int half = lane >> 4, lr = lane & 15;
  int m = row0 + lr;
  int n = col0 + lr;
  bool nok = (n < NCLS);
  const float* Arow = Xt + (long long)m * CH;
  v8f acc = {};
  for (int k0 = 0; k0 < CH; k0 += 4){
    int k = k0 + 2 * half;
    v2f a; a.x = Arow[k]; a.y = Arow[k + 1];
    v2f b;
    b.x = nok ? Wo[(long long)k * NCLS + n]       : 0.f;
    b.y = nok ? Wo[(long long)(k + 1) * NCLS + n] : 0.f;
    acc = __builtin_amdgcn_wmma_f32_16x16x4_f32(false, a, false, b, (short)0, acc, false, false);
  }
  if (nok){
    float bias = bo[n];
#pragma unroll
    for (int r = 0; r < 8; r++)
      Xp2[(long long)(row0 + r + 8 * half) * NCLS + n] = acc[r] + bias;
  }
}

// ---------------- per-pair attention logits, 8 heads of 64 ch (wave per pair) -----------
__global__ void k_pair_logits8(const float* __restrict__ F, const float* __restrict__ a,
                               const int* __restrict__ ids, float* __restrict__ S){
  int lane = threadIdx.x & 31;
  int wave = threadIdx.x >> 5;
  int p = blockIdx.x * 8 + wave;
  if (p >= NPAIR) return;
  long long base = (long long)ids[p] * CH;
#pragma unroll
  for (int h = 0; h < HEADS; h++){
    int c0 = h * 64 + lane;
    float t = F[base + c0] * a[h * 64 + lane] + F[base + c0 + 32] * a[h * 64 + lane + 32];
    t = wred32(t);
    if (lane == 0) S[(long long)p * HEADS + h] = leaky(t);
  }
}

// ---------------- per-pair logit, single head of 40 ch (thread per pair) ---------------
__global__ void k_pair_logit1(const float* __restrict__ F, const float* __restrict__ a,
                              const int* __restrict__ ids, float* __restrict__ S){
  int p = blockIdx.x * blockDim.x + threadIdx.x;
  if (p >= NPAIR) return;
  long long base = (long long)ids[p] * NCLS;
  float t = 0.f;
#pragma unroll
  for (int j = 0; j < NCLS; j++) t += F[base + j] * a[j];
  S[p] = leaky(t);
}

// ---------------- segment max via atomicMax on monotone-encoded floats -----------------
__global__ void k_segmax(const float* __restrict__ S, const int* __restrict__ seg,
                         unsigned* __restrict__ M, int Hn){
  long long i = (long long)blockIdx.x * blockDim.x + threadIdx.x;
  long long n = (long long)NPAIR * Hn;
  if (i >= n) return;
  int p = (int)(i / Hn), h = (int)(i % Hn);
  atomicMax(&M[(long long)seg[p] * Hn + h], fflip(S[i]));
}

// ---------------- ex = exp(s - max); den += ex -----------------------------------------
__global__ void k_exden(float* __restrict__ S, const int* __restrict__ seg,
                        const unsigned* __restrict__ M, float* __restrict__ D, int Hn){
  long long i = (long long)blockIdx.x * blockDim.x + threadIdx.x;
  long long n = (long long)NPAIR * Hn;
  if (i >= n) return;
  int p = (int)(i / Hn), h = (int)(i % Hn);
  long long si = (long long)seg[p] * Hn + h;
  float ex = expf(S[i] - funflip(M[si]));
  S[i] = ex;
  atomicAdd(&D[si], ex);
}

// ---------------- weighted scatter: O[dst, c] += (S/(D+eps)) * F[src, c] ---------------
__global__ void k_scatter(const float* __restrict__ F, const float* __restrict__ S,
                          const float* __restrict__ D, const int* __restrict__ src_ids,
                          const int* __restrict__ dst_ids, float* __restrict__ O,
                          int C, int Hn){
  long long i = (long long)blockIdx.x * blockDim.x + threadIdx.x;
  long long n = (long long)NPAIR * C;
  if (i >= n) return;
  int p = (int)(i / C), c = (int)(i % C);
  int h = c / (C / Hn);
  int s = src_ids[p], d = dst_ids[p];
  float al = S[(long long)p * Hn + h] / (D[(long long)d * Hn + h] + EPS);
  atomicAdd(&O[(long long)d * C + c], al * F[(long long)s * C + c]);
}

// ---------------- relu + p-laplacian-scaled sinusoid positional encoding ---------------
__global__ void k_posenc(float* __restrict__ Xt){
  int lane = threadIdx.x & 31;
  int wave = threadIdx.x >> 5;
  int v = blockIdx.x * 8 + wave;
  if (v >= NVERT) return;
  float* row = Xt + (long long)v * CH;
  float xr[16];
  float s = 0.f;
#pragma unroll
  for (int i = 0; i < 16; i++){
    float x = row[lane + 32 * i];
    x = x > 0.f ? x : 0.f;                      // relu (heads conv is_last=False)
    xr[i] = x; s += x;
  }
  s = wred32(s);                                // p_lap with P_NORM=1 on relu'd rows
  const float coef = -9.210340371976184f / 512.0f;   // -ln(10000)/C
  const float inv  = 0.04419417382415922f;           // 1/sqrt(512)
  float fv = (float)v;
#pragma unroll
  for (int i = 0; i < 16; i++){
    int c = lane + 32 * i;
    float ang = fv * expf((float)c * coef);
    float pe = (((c & 1) == 0) ? sinf(ang) : cosf(ang)) * inv;
    row[c] = xr[i] + s * pe;
  }
}

// ---------------- degree counts -------------------------------------------------------
__global__ void k_deg(const int* __restrict__ v_ids, const int* __restrict__ e_ids,
                      float* __restrict__ dv, float* __restrict__ de){
  int p = blockIdx.x * blockDim.x + threadIdx.x;
  if (p >= NPAIR) return;
  atomicAdd(&de[e_ids[p]], 1.0f);
  atomicAdd(&dv[v_ids[p]], 1.0f);
}

// ---------------- hspd: Xe[e] += Xt[v]  then  Xt[v] += Xe[e]/(de*dv) -------------------
__global__ void k_hspd_edge(const float* __restrict__ Xt, const int* __restrict__ v_ids,
                            const int* __restrict__ e_ids, float* __restrict__ Xe){
  long long i = (long long)blockIdx.x * blockDim.x + threadIdx.x;
  long long n = (long long)NPAIR * CH;
  if (i >= n) return;
  int p = (int)(i / CH), c = (int)(i % CH);
  atomicAdd(&Xe[(long long)e_ids[p] * CH + c], Xt[(long long)v_ids[p] * CH + c]);
}
__global__ void k_hspd_vert(const float* __restrict__ Xe, const int* __restrict__ v_ids,
                            const int* __restrict__ e_ids, const float* __restrict__ de,
                            const float* __restrict__ dv, float* __restrict__ Xt){
  long long i = (long long)blockIdx.x * blockDim.x + threadIdx.x;
  long long n = (long long)NPAIR * CH;
  if (i >= n) return;
  int p = (int)(i / CH), c = (int)(i % CH);
  int e = e_ids[p], v = v_ids[p];
  float wgt = 1.0f / (fmaxf(de[e], 1.f) * fmaxf(dv[v], 1.f));
  atomicAdd(&Xt[(long long)v * CH + c], wgt * Xe[(long long)e * CH + c]);
}

// ======================================================================================
extern "C" void kernel_launch(void* const* d_in, const int* in_sizes, int n_in,
                              void* d_out, int out_size, void* d_ws, size_t ws_size,
                              hipStream_t stream){
  (void)in_sizes; (void)n_in; (void)out_size; (void)ws_size;
  const float* X   = (const float*)d_in[0];
  const float* Wh  = (const float*)d_in[1];
  const float* bh  = (const float*)d_in[2];
  const float* avh = (const float*)d_in[3];
  const float* aeh = (const float*)d_in[4];
  const float* Wo  = (const float*)d_in[5];
  const float* bo  = (const float*)d_in[6];
  const float* avo = (const float*)d_in[7];
  const float* aeo = (const float*)d_in[8];
  const int* v_ids = (const int*)d_in[9];
  const int* e_ids = (const int*)d_in[10];
  float* out = (float*)d_out;

  // workspace carve-out (~160 MB), 256B aligned
  char* w = (char*)d_ws;
  size_t off = 0;
  auto carve = [&](size_t bytes) -> char* {
    char* p = w + off; off += (bytes + 255) & ~(size_t)255; return p;
  };
  float*    bufA = (float*)   carve((size_t)NVERT * CH * 4);     // Xp, later Xo/Xt
  float*    Ye   = (float*)   carve((size_t)NEDGE * CH * 4);     // Ye, later Xe, later Ye2
  float*    S    = (float*)   carve((size_t)NPAIR * HEADS * 4);  // pair logits / exp
  unsigned* eM   = (unsigned*)carve((size_t)NEDGE * HEADS * 4);
  float*    eD   = (float*)   carve((size_t)NEDGE * HEADS * 4);
  unsigned* vM   = (unsigned*)carve((size_t)NVERT * HEADS * 4);
  float*    vD   = (float*)   carve((size_t)NVERT * HEADS * 4);
  float*    de   = (float*)   carve((size_t)NEDGE * 4);
  float*    dv   = (float*)   carve((size_t)NVERT * 4);
  float*    Xp2  = (float*)   carve((size_t)NVERT * NCLS * 4);

  auto fill = [&](void* p, unsigned v, long long n){
    int blocks = (int)((n + 255) / 256);
    k_fill<<<blocks, 256, 0, stream>>>((unsigned*)p, v, n);
  };

  // ---- init accumulators ----
  fill(out, 0, (long long)NVERT * NCLS);
  fill(Ye,  0, (long long)NEDGE * CH);
  fill(eM, NEGINF_ENC, (long long)NEDGE * HEADS);
  fill(eD,  0, (long long)NEDGE * HEADS);
  fill(vM, NEGINF_ENC, (long long)NVERT * HEADS);
  fill(vD,  0, (long long)NVERT * HEADS);
  fill(de,  0, NEDGE);
  fill(dv,  0, NVERT);

  const long long nPH = (long long)NPAIR * HEADS;
  const long long nPC = (long long)NPAIR * CH;
  const long long nPJ = (long long)NPAIR * NCLS;

  // ---- stage 1: multi-head conv (all 8 heads fused on CH=512 layout) ----
  k_gemm1<<<(NVERT / 16) * (CH / 16) / 8, 256, 0, stream>>>(X, Wh, bh, bufA);
  k_pair_logits8<<<NPAIR / 8, 256, 0, stream>>>(bufA, avh, v_ids, S);
  k_segmax<<<(int)((nPH + 255) / 256), 256, 0, stream>>>(S, e_ids, eM, HEADS);
  k_exden <<<(int)((nPH + 255) / 256), 256, 0, stream>>>(S, e_ids, eM, eD, HEADS);
  k_scatter<<<(int)((nPC + 255) / 256), 256, 0, stream>>>(bufA, S, eD, v_ids, e_ids, Ye, CH, HEADS);
  fill(bufA, 0, (long long)NVERT * CH);                 // bufA becomes Xo accumulator
  k_pair_logits8<<<NPAIR / 8, 256, 0, stream>>>(Ye, aeh, e_ids, S);
  k_segmax<<<(int)((nPH + 255) / 256), 256, 0, stream>>>(S, v_ids, vM, HEADS);
  k_exden <<<(int)((nPH + 255) / 256), 256, 0, stream>>>(S, v_ids, vM, vD, HEADS);
  k_scatter<<<(int)((nPC + 255) / 256), 256, 0, stream>>>(Ye, S, vD, e_ids, v_ids, bufA, CH, HEADS);

  // ---- encodings: relu + positional, then two-hop structural smoothing ----
  k_posenc<<<NVERT / 8, 256, 0, stream>>>(bufA);
  k_deg<<<(NPAIR + 255) / 256, 256, 0, stream>>>(v_ids, e_ids, dv, de);
  fill(Ye, 0, (long long)NEDGE * CH);                   // Ye reused as Xe
  k_hspd_edge<<<(int)((nPC + 255) / 256), 256, 0, stream>>>(bufA, v_ids, e_ids, Ye);
  k_hspd_vert<<<(int)((nPC + 255) / 256), 256, 0, stream>>>(Ye, v_ids, e_ids, de, dv, bufA);

  // ---- stage 2: output conv (C=40, single head), is_last -> no relu ----
  k_gemm2<<<((NVERT / 16) * 3 + 7) / 8, 256, 0, stream>>>(bufA, Wo, bo, Xp2);
  fill(eM, NEGINF_ENC, NEDGE);
  fill(eD, 0, NEDGE);
  fill(vM, NEGINF_ENC, NVERT);
  fill(vD, 0, NVERT);
  fill(Ye, 0, (long long)NEDGE * NCLS);                 // Ye reused as Ye2
  k_pair_logit1<<<(NPAIR + 255) / 256, 256, 0, stream>>>(Xp2, avo, v_ids, S);
  k_segmax<<<(NPAIR + 255) / 256, 256, 0, stream>>>(S, e_ids, eM, 1);
  k_exden <<<(NPAIR + 255) / 256, 256, 0, stream>>>(S, e_ids, eM, eD, 1);
  k_scatter<<<(int)((nPJ + 255) / 256), 256, 0, stream>>>(Xp2, S, eD, v_ids, e_ids, Ye, NCLS, 1);
  k_pair_logit1<<<(NPAIR + 255) / 256, 256, 0, stream>>>(Ye, aeo, e_ids, S);
  k_segmax<<<(NPAIR + 255) / 256, 256, 0, stream>>>(S, v_ids, vM, 1);
  k_exden <<<(NPAIR + 255) / 256, 256, 0, stream>>>(S, v_ids, vM, vD, 1);
  k_scatter<<<(int)((nPJ + 255) / 256), 256, 0, stream>>>(Ye, S, vD, e_ids, v_ids, out, NCLS, 1);
}